// GMMDOTA_36747740184749
// MI455X (gfx1250) — compile-verified
//
#include <hip/hip_runtime.h>
#include <hip/hip_bf16.h>

#define BB   256
#define KK   1000
#define MM   4
#define DD   512
#define KM   4000            // K*M
#define KMD  2048000         // K*M*D

typedef float v2f __attribute__((ext_vector_type(2)));
typedef float v8f __attribute__((ext_vector_type(8)));

__device__ __forceinline__ v8f wmma_f32_k4(v2f a, v2f b, v8f c) {
    // D = A(16x4,f32) * B(4x16,f32) + C(16x16,f32)  -> v_wmma_f32_16x16x4_f32
    return __builtin_amdgcn_wmma_f32_16x16x4_f32(
        /*neg_a=*/false, a, /*neg_b=*/false, b,
        /*c_mod=*/(short)0, c, /*reuse_a=*/false, /*reuse_b=*/false);
}

// Hardware reciprocal (v_rcp_f32, ~1 ulp) — avoids the IEEE div-fixup chain.
__device__ __forceinline__ float hrcp(float x) {
    return __builtin_amdgcn_rcpf(x);
}

// ---------------------------------------------------------------------------
// Kernel 1: per-component constant  cterm[km] = log(pi) - 0.5*(sum mu^2/sigma
//           + sum log sigma_safe).  One 256-thread block per km row.
// ---------------------------------------------------------------------------
__global__ __launch_bounds__(256)
void gmm_prep_cterm(const float* __restrict__ mu, const float* __restrict__ Sigma,
                    const float* __restrict__ pi, float* __restrict__ cterm) {
    const int km  = blockIdx.x;
    const int tid = threadIdx.x;
    const float* mrow = mu    + (size_t)km * DD;
    const float* srow = Sigma + (size_t)km * DD;
    float acc = 0.f;
    for (int d = tid; d < DD; d += 256) {
        float s = fmaxf(srow[d], 1e-8f);
        float m = mrow[d];
        acc += logf(s) + m * m * hrcp(s);      // ldet + quad combined
    }
    __shared__ float red[256];
    red[tid] = acc;
    __syncthreads();
    for (int s = 128; s > 0; s >>= 1) {
        if (tid < s) red[tid] += red[tid + s];
        __syncthreads();
    }
    if (tid == 0)
        cterm[km] = logf(fmaxf(pi[km], 1e-10f)) - 0.5f * red[0];
}

// ---------------------------------------------------------------------------
// Kernel 2: xT[d*B + b] = x[b*D + d]; also zero sum_gamma.
// ---------------------------------------------------------------------------
__global__ __launch_bounds__(256)
void gmm_prep_xT(const float* __restrict__ x, float* __restrict__ xT,
                 float* __restrict__ sum_gamma) {
    const int idx = blockIdx.x * 256 + threadIdx.x;   // 0 .. 131071
    const int d = idx >> 8;
    const int b = idx & 255;
    xT[(size_t)d * BB + b] = x[(size_t)b * DD + d];
    if (idx < KM) sum_gamma[idx] = 0.f;
}

// ---------------------------------------------------------------------------
// Kernel 3: scores via two f32 WMMA GEMMs, softmax over M=4, gamma = y * r.
// One wave per 16(b) x 16(km) tile. 4000 waves, full 128-thread blocks.
// ---------------------------------------------------------------------------
__global__ __launch_bounds__(128)
void gmm_gemm1_gamma(const float* __restrict__ x, const float* __restrict__ y,
                     const float* __restrict__ mu, const float* __restrict__ Sigma,
                     const float* __restrict__ cterm,
                     float* __restrict__ gammaT, float* __restrict__ sum_gamma) {
    const int wave   = (blockIdx.x * blockDim.x + threadIdx.x) >> 5;
    const int lane   = threadIdx.x & 31;
    const int btile  = (wave & 15) << 4;     // b-tile varies fastest -> km rows reused
    const int kmtile = (wave >> 4) << 4;
    const int n  = lane & 15;                // fragment row / output column
    const int kb = (lane >> 4) << 1;         // contraction sub-offset {0,2}

    const float* xrow = x     + (size_t)(btile  + n) * DD + kb;
    const float* srow = Sigma + (size_t)(kmtile + n) * DD + kb;
    const float* mrow = mu    + (size_t)(kmtile + n) * DD + kb;

    v8f s1 = {}; v8f s2 = {};
    #pragma unroll 4
    for (int d = 0; d < DD; d += 4) {
        v2f xa = *(const v2f*)(xrow + d);          // A-frag (x)
        v2f sg = *(const v2f*)(srow + d);
        v2f mv = *(const v2f*)(mrow + d);
        v2f inv;
        inv.x = hrcp(fmaxf(sg.x, 1e-8f));
        inv.y = hrcp(fmaxf(sg.y, 1e-8f));
        v2f xsq = xa * xa;                         // A-frag (x^2), same layout
        v2f mi  = mv * inv;                        // B-frag (mu*inv_sigma)
        s1 = wmma_f32_k4(xsq, inv, s1);            // sum_d x^2 * inv_sigma
        s2 = wmma_f32_k4(xa,  mi,  s2);            // sum_d x * mu * inv_sigma
    }

    const float ct  = cterm[kmtile + n];
    const int   hi  = lane >> 4;
    const int   kcl = (kmtile + n) >> 2;           // class index k = km/4
    float lsum = 0.f;
    #pragma unroll
    for (int i = 0; i < 8; ++i) {
        // score = log_pi + log_l = -0.5*S1 + S2 + cterm
        float sc = -0.5f * s1[i] + s2[i] + ct;
        // softmax over M=4 (4 consecutive lanes)
        float mx = fmaxf(sc, __shfl_xor(sc, 1, 4));
        mx = fmaxf(mx, __shfl_xor(mx, 2, 4));
        float e  = __expf(sc - mx);
        float sm = e + __shfl_xor(e, 1, 4);
        sm += __shfl_xor(sm, 2, 4);
        const int brow = btile + i + (hi << 3);
        float g = y[(size_t)brow * KK + kcl] * (e * hrcp(sm));   // sm >= 1
        gammaT[(size_t)(kmtile + n) * BB + brow] = g;
        lsum += g;
    }
    atomicAdd(&sum_gamma[kmtile + n], lsum);
}

// ---------------------------------------------------------------------------
// Kernel 4: weighted_x / weighted_xx via two f32 WMMA GEMMs over B, fused
// new_mu / new_Sigma epilogue. One wave per 16(km) x 16(d) tile. 8000 waves.
// ---------------------------------------------------------------------------
__global__ __launch_bounds__(128)
void gmm_gemm2_out(const float* __restrict__ gammaT, const float* __restrict__ xT,
                   const float* __restrict__ mu, const float* __restrict__ Sigma,
                   const float* __restrict__ Cc, const float* __restrict__ sum_gamma,
                   float* __restrict__ out) {
    const int wave   = (blockIdx.x * blockDim.x + threadIdx.x) >> 5;
    const int lane   = threadIdx.x & 31;
    const int kmtile = (wave >> 5) << 4;           // 250 km-tiles
    const int dtile  = (wave & 31) << 4;           // 32 d-tiles
    const int n  = lane & 15;
    const int kb = (lane >> 4) << 1;

    const float* grow = gammaT + (size_t)(kmtile + n) * BB + kb;  // A-frag rows
    const float* xrow = xT     + (size_t)(dtile  + n) * BB + kb;  // B-frag cols

    v8f w1 = {}; v8f w2 = {};
    #pragma unroll 8
    for (int b0 = 0; b0 < BB; b0 += 4) {
        v2f g  = *(const v2f*)(grow + b0);
        v2f xa = *(const v2f*)(xrow + b0);
        v2f xs = xa * xa;
        w1 = wmma_f32_k4(g, xa, w1);               // sum_b gamma * x
        w2 = wmma_f32_k4(g, xs, w2);               // sum_b gamma * x^2
    }

    const int hi = lane >> 4;
    const int d  = dtile + n;
    #pragma unroll
    for (int i = 0; i < 8; ++i) {
        const int km = kmtile + i + (hi << 3);
        float Cv  = Cc[km];
        float sg  = sum_gamma[km];
        float muv = mu[(size_t)km * DD + d];
        float sv  = Sigma[(size_t)km * DD + d];
        float rden = hrcp(fmaxf(Cv + sg, 1e-10f));
        float nm  = (Cv * muv + w1[i]) * rden;
        float wds = w2[i] - 2.f * muv * w1[i] + muv * muv * sg;
        float ns  = fmaxf((Cv * sv + wds) * rden, 1e-8f);
        out[(size_t)km * DD + d] = nm;                      // new_mu plane
        out[(size_t)KMD + (size_t)km * DD + d] = ns;        // new_Sigma plane
    }
}

// ---------------------------------------------------------------------------
extern "C" void kernel_launch(void* const* d_in, const int* in_sizes, int n_in,
                              void* d_out, int out_size, void* d_ws, size_t ws_size,
                              hipStream_t stream) {
    const float* x     = (const float*)d_in[0];   // (B,D)
    const float* y     = (const float*)d_in[1];   // (B,K)
    const float* mu    = (const float*)d_in[2];   // (K,M,D)
    const float* Sigma = (const float*)d_in[3];   // (K,M,D)
    const float* pi    = (const float*)d_in[4];   // (K,M)
    const float* Cc    = (const float*)d_in[5];   // (K,M)
    float* out = (float*)d_out;                   // (2,K,M,D)

    float* ws        = (float*)d_ws;
    float* cterm     = ws;                        // 4000  (padded 4096)
    float* sum_gamma = cterm + 4096;              // 4000  (padded 4096)
    float* xT        = sum_gamma + 4096;          // 512*256 = 131072
    float* gammaT    = xT + 131072;               // 4000*256 = 1,024,000
    (void)in_sizes; (void)n_in; (void)out_size; (void)ws_size;

    gmm_prep_cterm<<<KM, 256, 0, stream>>>(mu, Sigma, pi, cterm);
    gmm_prep_xT<<<512, 256, 0, stream>>>(x, xT, sum_gamma);
    gmm_gemm1_gamma<<<1000, 128, 0, stream>>>(x, y, mu, Sigma, cterm,
                                              gammaT, sum_gamma);
    gmm_gemm2_out<<<2000, 128, 0, stream>>>(gammaT, xT, mu, Sigma, Cc,
                                            sum_gamma, out);
}